// OneLayerGCNWithGlobalAdg_32444182954834
// MI455X (gfx1250) — compile-verified
//
#include <hip/hip_runtime.h>
#include <hip/hip_bf16.h>

typedef __attribute__((ext_vector_type(2))) float v2f;
typedef __attribute__((ext_vector_type(8))) float v8f;

#define NN   262144
#define EE   4194304
#define FIN  128
#define FOUT 64
#define GG   65536
#define APITCH 132   // padded LDS pitch for A tile (floats)

// Native fp32 global atomic add (non-returning), device scope.
// global_atomic_add_f32 vaddr(64b VGPR pair), vdata, off
__device__ __forceinline__ void gadd_f32(float* p, float v) {
    unsigned long long a = (unsigned long long)p;
    asm volatile("global_atomic_add_f32 %0, %1, off scope:SCOPE_DEV"
                 :: "v"(a), "v"(v) : "memory");
}

// ---------------- Kernel 0: zero h ----------------
__global__ __launch_bounds__(256)
void zero_kernel(float4* __restrict__ p, long n4) {
    long i = (long)blockIdx.x * blockDim.x + threadIdx.x;
    long stride = (long)gridDim.x * blockDim.x;
    float4 z = make_float4(0.f, 0.f, 0.f, 0.f);
    for (; i < n4; i += stride) p[i] = z;
}

// ---------------- Kernel 1: tmp = feat @ weight (WMMA f32 16x16x4) ----------------
// grid = N/16 blocks, 128 threads (4 waves). Wave w computes 16x16 tile at cols [16w,16w+16).
__global__ __launch_bounds__(128)
void gemm_feat_weight(const float* __restrict__ feat,
                      const float* __restrict__ weight,
                      float* __restrict__ tmp) {
    __shared__ float As[16 * APITCH];
    const int tid  = threadIdx.x;
    const int wave = tid >> 5;
    const int lane = tid & 31;
    const long rowBase = (long)blockIdx.x * 16;

    // Cooperative load of A tile: 16 rows x 128 floats; 8 threads/row, 16 floats each.
    {
        const int r  = tid >> 3;
        const int c0 = (tid & 7) * 16;
        const float4* src = (const float4*)(feat + (rowBase + r) * FIN + c0);
        float4 x0 = src[0], x1 = src[1], x2 = src[2], x3 = src[3];
        float4* dst = (float4*)(As + r * APITCH + c0);
        dst[0] = x0; dst[1] = x1; dst[2] = x2; dst[3] = x3;
    }
    __syncthreads();

    // Preload B fragments (weight is tiny + L2 resident; 32 steps x v2f = 64 VGPRs).
    const int nBase = wave * 16;
    const int n  = lane & 15;
    const int kk = (lane >> 4) * 2;   // lanes 0-15 -> K+0/1, lanes 16-31 -> K+2/3
    v2f bfrag[32];
#pragma unroll
    for (int s = 0; s < 32; ++s) {
        const int k = s * 4 + kk;
        bfrag[s].x = weight[k       * FOUT + nBase + n];
        bfrag[s].y = weight[(k + 1) * FOUT + nBase + n];
    }

    const int m = lane & 15;
    v8f acc = {};
#pragma unroll
    for (int s = 0; s < 32; ++s) {
        const int k = s * 4 + kk;
        v2f afrag;
        afrag.x = As[m * APITCH + k];
        afrag.y = As[m * APITCH + k + 1];
        acc = __builtin_amdgcn_wmma_f32_16x16x4_f32(
            /*neg_a=*/false, afrag, /*neg_b=*/false, bfrag[s],
            /*c_mod=*/(short)0, acc, /*reuse_a=*/false, /*reuse_b=*/false);
    }

    // D layout: VGPR v -> row m = v + (lane>>4)*8, col = nBase + (lane&15)
    const int mOff = (lane >> 4) * 8;
    const int col  = nBase + (lane & 15);
#pragma unroll
    for (int v = 0; v < 8; ++v) {
        tmp[(rowBase + mOff + v) * FOUT + col] = acc[v];
    }
}

// ---------------- Kernel 2: scatter-add edge messages into h ----------------
// One wave per 8 edges; 32 lanes cover 64 features (2 per lane). Anchor sources skipped (zero msg).
__global__ __launch_bounds__(256)
void scatter_edges(const float* __restrict__ tmp,
                   const int* __restrict__ esrc,
                   const int* __restrict__ edst,
                   const float* __restrict__ ew,
                   float* __restrict__ h, int nEdges) {
    const int lane   = threadIdx.x & 31;
    const int waveId = blockIdx.x * (blockDim.x >> 5) + (threadIdx.x >> 5);
    const int e0 = waveId * 8;
#pragma unroll
    for (int i = 0; i < 8; ++i) {
        const int e = e0 + i;
        if (e >= nEdges) return;
        const int src = esrc[e];
        if ((src & 3) == 0) continue;   // anchor row: in_feat row is zero
        const int dst = edst[e];
        const float w = ew[e];
        const float v0 = tmp[(long)src * FOUT + lane]      * w;
        const float v1 = tmp[(long)src * FOUT + lane + 32] * w;
        float* hp = h + (long)dst * FOUT + lane;
        gadd_f32(hp,      v0);
        gadd_f32(hp + 32, v1);
    }
}

// ---------------- Kernel 3: prelu+pool+matvecs+l2norm ----------------
// One wave per group g (4 rows). 8 waves / block.
__global__ __launch_bounds__(256)
void finalize_kernel(const float* __restrict__ tmp,
                     const float* __restrict__ h,
                     const float* __restrict__ bias,
                     const float* __restrict__ prelu_a,
                     const float* __restrict__ w_sub, const float* __restrict__ b_sub,
                     const float* __restrict__ w_gcn, const float* __restrict__ b_gcn,
                     float* __restrict__ out) {
    __shared__ float wsub_s[64 * 65];
    __shared__ float wgcn_s[64 * 65];
    __shared__ float pooled_s[8][64];
    __shared__ float gcnv_s[8][64];

    const int tid = threadIdx.x;
    for (int i = tid; i < 64 * 64; i += 256) {
        const int r = i >> 6, c = i & 63;
        wsub_s[r * 65 + c] = w_sub[i];
        wgcn_s[r * 65 + c] = w_gcn[i];
    }
    __syncthreads();

    const int wave = tid >> 5, lane = tid & 31;
    const int g = blockIdx.x * 8 + wave;
    const float a  = prelu_a[0];
    const int c0 = lane, c1 = lane + 32;
    const float b0 = bias[c0], b1 = bias[c1];

    float p0 = 0.f, p1 = 0.f, g0 = 0.f, g1 = 0.f;
#pragma unroll
    for (int r = 0; r < 4; ++r) {
        const long base = ((long)g * 4 + r) * FOUT;
        float x0 = h[base + c0] + b0;
        float x1 = h[base + c1] + b1;
        x0 = x0 >= 0.f ? x0 : a * x0;
        x1 = x1 >= 0.f ? x1 : a * x1;
        p0 += x0; p1 += x1;
        if (r == 0) { g0 = x0; g1 = x1; }
    }
    p0 *= 0.25f; p1 *= 0.25f;
    pooled_s[wave][c0] = p0; pooled_s[wave][c1] = p1;
    gcnv_s[wave][c0]   = g0; gcnv_s[wave][c1]   = g1;

    // anchor_out = prelu(tmp[4g] + bias)
    const long abase = (long)g * 4 * FOUT;
    float an0 = tmp[abase + c0] + b0; an0 = an0 >= 0.f ? an0 : a * an0;
    float an1 = tmp[abase + c1] + b1; an1 = an1 >= 0.f ? an1 : a * an1;

    __syncthreads();

    // y[j] = b[j] + sum_c v[c] * W[j][c] ; lane computes j = lane and lane+32
    float ys0 = b_sub[c0], ys1 = b_sub[c1];
    float yg0 = b_gcn[c0], yg1 = b_gcn[c1];
#pragma unroll 8
    for (int c = 0; c < 64; ++c) {
        const float pc = pooled_s[wave][c];
        const float gc = gcnv_s[wave][c];
        ys0 = fmaf(pc, wsub_s[c0 * 65 + c], ys0);
        ys1 = fmaf(pc, wsub_s[c1 * 65 + c], ys1);
        yg0 = fmaf(gc, wgcn_s[c0 * 65 + c], yg0);
        yg1 = fmaf(gc, wgcn_s[c1 * 65 + c], yg1);
    }

    float ss = ys0 * ys0 + ys1 * ys1;
    float sg = yg0 * yg0 + yg1 * yg1;
    float sa = an0 * an0 + an1 * an1;
#pragma unroll
    for (int off = 16; off > 0; off >>= 1) {
        ss += __shfl_xor(ss, off, 32);
        sg += __shfl_xor(sg, off, 32);
        sa += __shfl_xor(sa, off, 32);
    }
    const float EPSV = 1e-12f;
    const float ns = 1.f / fmaxf(sqrtf(ss), EPSV);
    const float ng = 1.f / fmaxf(sqrtf(sg), EPSV);
    const float na = 1.f / fmaxf(sqrtf(sa), EPSV);

    const long ob = (long)g * FOUT;
    const long OG = (long)GG * FOUT;
    out[ob + c0]          = ys0 * ns; out[ob + c1]          = ys1 * ns;
    out[OG + ob + c0]     = an0 * na; out[OG + ob + c1]     = an1 * na;
    out[2 * OG + ob + c0] = yg0 * ng; out[2 * OG + ob + c1] = yg1 * ng;
}

extern "C" void kernel_launch(void* const* d_in, const int* in_sizes, int n_in,
                              void* d_out, int out_size, void* d_ws, size_t ws_size,
                              hipStream_t stream) {
    const float* feat    = (const float*)d_in[0];
    const int*   esrc    = (const int*)  d_in[1];
    const int*   edst    = (const int*)  d_in[2];
    const float* ew      = (const float*)d_in[3];
    const float* weight  = (const float*)d_in[4];
    const float* bias    = (const float*)d_in[5];
    const float* prelu_a = (const float*)d_in[6];
    const float* w_sub   = (const float*)d_in[7];
    const float* b_sub   = (const float*)d_in[8];
    const float* w_gcn   = (const float*)d_in[9];
    const float* b_gcn   = (const float*)d_in[10];
    float* out = (float*)d_out;

    float* tmp = (float*)d_ws;                    // N*64 floats
    float* h   = tmp + (size_t)NN * FOUT;         // N*64 floats

    zero_kernel<<<2048, 256, 0, stream>>>((float4*)h, (long)NN * FOUT / 4);
    gemm_feat_weight<<<NN / 16, 128, 0, stream>>>(feat, weight, tmp);
    scatter_edges<<<EE / 64, 256, 0, stream>>>(tmp, esrc, edst, ew, h, EE);
    finalize_kernel<<<GG / 8, 256, 0, stream>>>(tmp, h, bias, prelu_a,
                                                w_sub, b_sub, w_gcn, b_gcn, out);
}